// CapsuleFC_3805341024308
// MI455X (gfx1250) — compile-verified
//
#include <hip/hip_runtime.h>

typedef float v2f __attribute__((ext_vector_type(2)));
typedef float v8f __attribute__((ext_vector_type(8)));

#define B_CAP   16
#define NIN     256
#define DIN     128
#define NOUT    128
#define DOUT    64
#define COLS    (NOUT * DOUT)      // 8192
#define NGROUPS 8
#define NPER    (NIN / NGROUPS)    // 32
#define A_STRIDE 132               // padded LDS row stride (dwords): conflict-free b64 reads

// ---------------------------------------------------------------------------
// Phase 0: na[b] = mean(current_act[b,:]); next_act output = broadcast of na.
// ---------------------------------------------------------------------------
__global__ __launch_bounds__(256)
void act_mean_kernel(const float* __restrict__ act, float* __restrict__ na,
                     float* __restrict__ out_next_act) {
    __shared__ float red[256];
    const int b = blockIdx.x;
    const int t = threadIdx.x;
    red[t] = act[b * NIN + t];
    __syncthreads();
    for (int s = 128; s > 0; s >>= 1) {
        if (t < s) red[t] += red[t + s];
        __syncthreads();
    }
    const float m = red[0] * (1.0f / NIN);
    if (t < NOUT) out_next_act[b * NOUT + t] = m;
    if (t == 0) na[b] = m;
}

// ---------------------------------------------------------------------------
// Phase 1: votes[b,n,m,d] = sum_a x[b,n,a] * W[n,a,m,d]  via f32 WMMA.
// Grid: (64 col-chunks of 128, 8 n-groups). Block: 256 thr = 8 waves.
// Wave w handles cols [chunk*128 + w*16, +16) ; loops its 32 n values.
// Also accumulates partial ncv sums per n-group (race-free slots).
// ---------------------------------------------------------------------------
__global__ __launch_bounds__(256)
void votes_kernel(const float* __restrict__ x, const float* __restrict__ W,
                  float* __restrict__ votes, float* __restrict__ partial) {
    __shared__ float Alds[16 * A_STRIDE];      // 8448 B: x_n tile, padded rows

    const int lane = threadIdx.x & 31;
    const int wave = threadIdx.x >> 5;                 // 0..7
    const int c0   = blockIdx.x * 128 + wave * 16;     // this wave's 16-col base
    const int g    = blockIdx.y;                       // n-group 0..7
    const int half = lane >> 4;                        // 0 | 1
    const int l16  = lane & 15;                        // col-in-tile / A-row

    // cooperative A-load mapping: thread t loads x[b = t/16, n, 8*(t%16) ..+7]
    const int lb = threadIdx.x >> 4;                   // 0..15
    const int lk = (threadIdx.x & 15) * 8;             // 0..120

    v8f acc = {0.f, 0.f, 0.f, 0.f, 0.f, 0.f, 0.f, 0.f};

    for (int i = 0; i < NPER; ++i) {
        const int n = g * NPER + i;

        __syncthreads();                               // protect previous A tile
        {   // stage x[:, n, :] (16 x 128 f32) into LDS, wide loads
            const float4* src = (const float4*)(x + ((size_t)lb * NIN + n) * DIN + lk);
            float4 p0 = src[0];
            float4 p1 = src[1];
            float* dst = &Alds[lb * A_STRIDE + lk];
            *(float4*)(dst)     = p0;
            *(float4*)(dst + 4) = p1;
        }
        __syncthreads();

        const float* Wn = W + (size_t)n * DIN * COLS + c0 + l16;

        // software-prefetch next n's W tile (rows p*32+lane, same cols)
        if (i + 1 < NPER) {
            const float* Wnx = Wn + (size_t)DIN * COLS;
#pragma unroll
            for (int p = 0; p < 4; ++p)
                __builtin_prefetch(Wnx + (size_t)(p * 32 + lane) * COLS, 0, 0);
        }

        v8f c = {0.f, 0.f, 0.f, 0.f, 0.f, 0.f, 0.f, 0.f};
#pragma unroll
        for (int kk = 0; kk < 32; ++kk) {
            const int krow = kk * 4 + half * 2;
            // A frag (16x4 f32 layout): v0 = K {0|2}, v1 = K {1|3}
            v2f a = *(const v2f*)&Alds[l16 * A_STRIDE + krow];
            // B frag (4x16 f32, mirrored layout): lane col = c0+l16
            v2f b;
            b.x = Wn[(size_t)krow * COLS];
            b.y = Wn[(size_t)(krow + 1) * COLS];
            c = __builtin_amdgcn_wmma_f32_16x16x4_f32(
                    false, a, false, b, (short)0, c, false, false);
        }
        acc += c;

        // store votes tile: votes[(n*16 + b)*COLS + col]
        float* vt = votes + (size_t)n * 16 * COLS + c0 + l16;
#pragma unroll
        for (int r = 0; r < 8; ++r) {
            const int bb = r + half * 8;
            vt[(size_t)bb * COLS] = c[r];
        }
    }

    // per-group partial ncv: partial[(g*16 + b)*COLS + col] (exactly one writer)
    float* pp = partial + (size_t)g * 16 * COLS + c0 + l16;
#pragma unroll
    for (int r = 0; r < 8; ++r) {
        const int bb = r + half * 8;
        pp[(size_t)bb * COLS] = acc[r];
    }
}

// ---------------------------------------------------------------------------
// Phase 2: ncv[b,col] = (1/NOUT) * sum_g partial[g,b,col]   (deterministic)
// ---------------------------------------------------------------------------
__global__ __launch_bounds__(256)
void ncv_reduce_kernel(const float* __restrict__ partial, float* __restrict__ ncv) {
    const int idx = blockIdx.x * 256 + threadIdx.x;    // over b*COLS + col
    const int b   = idx >> 13;
    const int col = idx & (COLS - 1);
    float s = 0.f;
#pragma unroll
    for (int g = 0; g < NGROUPS; ++g)
        s += partial[((size_t)(g * 16 + b)) * COLS + col];
    ncv[idx] = s * (1.0f / NOUT);
}

// ---------------------------------------------------------------------------
// Phase 3: scores -> softmax -> qk.  One block per (n, b); thread m in [0,128).
// ---------------------------------------------------------------------------
__global__ __launch_bounds__(128)
void qk_kernel(const float* __restrict__ votes, const float* __restrict__ ncv,
               const float* __restrict__ na, float* __restrict__ qk_out) {
    const int n = blockIdx.x;
    const int b = blockIdx.y;
    const int m = threadIdx.x;

    const float4* vp = (const float4*)(votes + ((size_t)n * 16 + b) * COLS + m * DOUT);
    const float4* cp = (const float4*)(ncv + (size_t)b * COLS + m * DOUT);
    float s = 0.f;
#pragma unroll
    for (int j = 0; j < 16; ++j) {
        const float4 v = vp[j];
        const float4 c = cp[j];
        s += v.x * c.x + v.y * c.y + v.z * c.z + v.w * c.w;
    }
    s *= 0.125f;                                       // 1/sqrt(DOUT)

    __shared__ float red[128];
    red[m] = s; __syncthreads();
    for (int st = 64; st > 0; st >>= 1) {
        if (m < st) red[m] = fmaxf(red[m], red[m + st]);
        __syncthreads();
    }
    const float mx = red[0]; __syncthreads();

    const float e = __expf(s - mx);
    red[m] = e; __syncthreads();
    for (int st = 64; st > 0; st >>= 1) {
        if (m < st) red[m] += red[m + st];
        __syncthreads();
    }
    const float esum = red[0]; __syncthreads();

    const float w = (e / esum) * na[b];
    red[m] = w; __syncthreads();
    for (int st = 64; st > 0; st >>= 1) {
        if (m < st) red[m] += red[m + st];
        __syncthreads();
    }
    const float denom = red[0] + 1e-10f;

    qk_out[((size_t)b * NIN + n) * NOUT + m] = w / denom;
}

// ---------------------------------------------------------------------------
// Phase 4: next_capsule_value[b,m,d] = sum_n qk[b,n,m]*act[b,n]*votes[b,n,m,d]
// One block per (m, b); thread d in [0,64). Deterministic n loop.
// ---------------------------------------------------------------------------
__global__ __launch_bounds__(64)
void capsule_reduce_kernel(const float* __restrict__ votes,
                           const float* __restrict__ qk,
                           const float* __restrict__ act,
                           float* __restrict__ out_ncval) {
    const int m = blockIdx.x;
    const int b = blockIdx.y;
    const int d = threadIdx.x;

    const float* vbase = votes + (size_t)b * COLS + m * DOUT + d;  // + n*16*COLS
    const float* qbase = qk + (size_t)b * NIN * NOUT + m;          // + n*NOUT
    const float* abase = act + b * NIN;

    float acc = 0.f;
#pragma unroll 4
    for (int n = 0; n < NIN; ++n) {
        const float w = qbase[(size_t)n * NOUT] * abase[n];
        acc += w * vbase[(size_t)n * 16 * COLS];
    }
    out_ncval[((size_t)b * NOUT + m) * DOUT + d] = acc;
}

// ---------------------------------------------------------------------------
extern "C" void kernel_launch(void* const* d_in, const int* in_sizes, int n_in,
                              void* d_out, int out_size, void* d_ws, size_t ws_size,
                              hipStream_t stream) {
    const float* x   = (const float*)d_in[0];   // [16,256,128]
    const float* act = (const float*)d_in[1];   // [16,256]
    const float* W   = (const float*)d_in[2];   // [256,128,128,64]
    // d_in[3] = num_iter (unused by reference body)

    float* out          = (float*)d_out;
    float* out_ncval    = out;                                    // 16*128*64
    float* out_next_act = out + (size_t)B_CAP * NOUT * DOUT;      // 16*128
    float* out_qk       = out_next_act + (size_t)B_CAP * NOUT;    // 16*256*128

    float* ws      = (float*)d_ws;
    float* votes   = ws;                                          // 33,554,432 f
    float* partial = votes + (size_t)B_CAP * NIN * COLS;          // 1,048,576 f
    float* ncv     = partial + (size_t)NGROUPS * B_CAP * COLS;    // 131,072 f
    float* na      = ncv + (size_t)B_CAP * COLS;                  // 16 f

    act_mean_kernel<<<dim3(B_CAP), dim3(256), 0, stream>>>(act, na, out_next_act);
    votes_kernel<<<dim3(64, NGROUPS), dim3(256), 0, stream>>>(x, W, votes, partial);
    ncv_reduce_kernel<<<dim3((B_CAP * COLS) / 256), dim3(256), 0, stream>>>(partial, ncv);
    qk_kernel<<<dim3(NIN, B_CAP), dim3(128), 0, stream>>>(votes, ncv, na, out_qk);
    capsule_reduce_kernel<<<dim3(NOUT, B_CAP), dim3(64), 0, stream>>>(votes, out_qk, act, out_ncval);
}